// BinomialTreeLayer_69501160784612
// MI455X (gfx1250) — compile-verified
//
#include <hip/hip_runtime.h>
#include <hip/hip_bf16.h>

typedef float v2f __attribute__((ext_vector_type(2)));
typedef float v8f __attribute__((ext_vector_type(8)));

#define BATCH 32
#define IN_CH 512
#define VDIM 16
#define HW 4096

// ---------------------------------------------------------------------------
// Kernel 1: root conv  out(b,16,HW) = root_w(16,512) @ features(b,512,HW) + root_b
// f32 WMMA 16x16x4. One block = 1 batch x 512 pixels, 8 waves, each wave a
// 16x64 tile with 4 accumulators.
// ---------------------------------------------------------------------------
__global__ __launch_bounds__(256)
void root_conv_wmma(const float* __restrict__ feat, const float* __restrict__ rw,
                    const float* __restrict__ rb, float* __restrict__ out)
{
    __shared__ float w[VDIM * IN_CH];              // 32 KB: root_w row-major
    {
        const float4* s4 = (const float4*)rw;
        float4* d4 = (float4*)w;
        for (int i = threadIdx.x; i < VDIM * IN_CH / 4; i += 256) d4[i] = s4[i];
    }
    __syncthreads();

    const int b      = blockIdx.x >> 3;
    const int pblock = (blockIdx.x & 7) * 512;
    const int wave   = threadIdx.x >> 5;
    const int lane   = threadIdx.x & 31;
    const int hi     = lane >> 4;                  // half-wave select
    const int mn     = lane & 15;                  // M (for A) == N (for B)
    const int pwave  = pblock + wave * 64;

    const float* fb = feat + (size_t)b * IN_CH * HW;

    v8f acc0 = {}, acc1 = {}, acc2 = {}, acc3 = {};
    for (int k = 0; k < IN_CH; k += 4) {
        const int ch = k + 2 * hi;
        // A fragment: row mn of root_w, cols ch, ch+1 (8B aligned in LDS)
        v2f a = *(const v2f*)&w[mn * IN_CH + ch];
        const float* fp = fb + (size_t)ch * HW + pwave + mn;
        v2f b0; b0.x = fp[0];  b0.y = fp[HW];
        v2f b1; b1.x = fp[16]; b1.y = fp[HW + 16];
        v2f b2; b2.x = fp[32]; b2.y = fp[HW + 32];
        v2f b3; b3.x = fp[48]; b3.y = fp[HW + 48];
        acc0 = __builtin_amdgcn_wmma_f32_16x16x4_f32(false, a, false, b0, (short)0, acc0, false, false);
        acc1 = __builtin_amdgcn_wmma_f32_16x16x4_f32(false, a, false, b1, (short)0, acc1, false, false);
        acc2 = __builtin_amdgcn_wmma_f32_16x16x4_f32(false, a, false, b2, (short)0, acc2, false, false);
        acc3 = __builtin_amdgcn_wmma_f32_16x16x4_f32(false, a, false, b3, (short)0, acc3, false, false);
    }

    float* ob = out + (size_t)b * VDIM * HW;
    #pragma unroll
    for (int r = 0; r < 8; ++r) {
        const int oc = r + 8 * hi;
        const float bias = rb[oc];
        float* op = ob + (size_t)oc * HW + pwave + mn;
        op[0]  = acc0[r] + bias;
        op[16] = acc1[r] + bias;
        op[32] = acc2[r] + bias;
        op[48] = acc3[r] + bias;
    }
}

// ---------------------------------------------------------------------------
// Kernel 2: spatial mean of feats0 per (batch, channel)
// ---------------------------------------------------------------------------
__global__ __launch_bounds__(256)
void mean_kernel(const float* __restrict__ f0, float* __restrict__ mean)
{
    const int b = blockIdx.x >> 4;
    const int c = blockIdx.x & 15;
    const float* row = f0 + ((size_t)b * VDIM + c) * HW;
    float s = 0.f;
    for (int i = threadIdx.x; i < HW; i += 256) s += row[i];
    __shared__ float red[256];
    red[threadIdx.x] = s;
    __syncthreads();
    for (int st = 128; st > 0; st >>= 1) {
        if (threadIdx.x < st) red[threadIdx.x] += red[threadIdx.x + st];
        __syncthreads();
    }
    if (threadIdx.x == 0) mean[b * VDIM + c] = red[0] * (1.0f / HW);
}

// ---------------------------------------------------------------------------
// Kernel 3: whole tree on pooled means (linear => exact), logits, argmax,
// and composition of the selected path into per-batch affine (M_b, v_b).
// One thread per batch.
// ---------------------------------------------------------------------------
__global__ void tree_kernel(const float* __restrict__ mean,
                            const float* __restrict__ lw, const float* __restrict__ lb,
                            float* __restrict__ Mb, float* __restrict__ vb,
                            float* __restrict__ dout)
{
    const int b = threadIdx.x;
    if (b >= BATCH) return;
    const int LOFF[4] = {0, 2, 5, 9};

    float cur[5][16], nxt[5][16];
    int cp[5][5];
    for (int c = 0; c < 16; ++c) cur[0][c] = mean[b * 16 + c];

    for (int level = 1; level <= 4; ++level) {
        const int off = LOFF[level - 1];
        for (int node = 0; node <= level; ++node) {
            int pl = (node - 1 < 0) ? 0 : node - 1;
            int pr = (node < level - 1) ? node : level - 1;
            int chosen = pl;
            if (pl != pr) {
                float ln = 0.f, rn = 0.f;
                for (int c = 0; c < 16; ++c) {
                    ln += cur[pl][c] * cur[pl][c];
                    rn += cur[pr][c] * cur[pr][c];
                }
                chosen = (rn > ln) ? pr : pl;   // sel = (rn > ln)
            }
            cp[level][node] = chosen;
            const float* W  = lw + (size_t)(off + node) * 256;
            const float* bi = lb + (size_t)(off + node) * 16;
            for (int o = 0; o < 16; ++o) {
                float s = bi[o];
                for (int c = 0; c < 16; ++c) s += W[o * 16 + c] * cur[chosen][c];
                nxt[node][o] = s;
            }
        }
        for (int node = 0; node <= level; ++node)
            for (int c = 0; c < 16; ++c) cur[node][c] = nxt[node][c];
    }

    // class logits + argmax (first max wins, like jnp.argmax)
    int sel = 0; float best = -1.f;
    for (int n = 0; n < 5; ++n) {
        float s = 0.f;
        for (int c = 0; c < 16; ++c) s += cur[n][c] * cur[n][c];
        float l = sqrtf(s);
        dout[b * 5 + n] = l;
        if (l > best) { best = l; sel = n; }
    }
    dout[192 + b] = (float)sel;

    // walk back the selected leaf's parent path
    int path[5]; path[4] = sel;
    for (int level = 4; level >= 1; --level) path[level - 1] = cp[level][path[level]];

    // compose affine map along the path: x_leaf = M x_root + v
    float M[256], v[16], nm[256], nv[16];
    for (int i = 0; i < 256; ++i) M[i] = 0.f;
    for (int i = 0; i < 16; ++i) { M[i * 16 + i] = 1.f; v[i] = 0.f; }
    for (int level = 1; level <= 4; ++level) {
        const int off  = LOFF[level - 1];
        const int node = path[level];
        const float* W  = lw + (size_t)(off + node) * 256;
        const float* bi = lb + (size_t)(off + node) * 16;
        for (int o = 0; o < 16; ++o) {
            float s = bi[o];
            for (int c = 0; c < 16; ++c) s += W[o * 16 + c] * v[c];
            nv[o] = s;
            for (int j = 0; j < 16; ++j) {
                float t = 0.f;
                for (int c = 0; c < 16; ++c) t += W[o * 16 + c] * M[c * 16 + j];
                nm[o * 16 + j] = t;
            }
        }
        for (int i = 0; i < 256; ++i) M[i] = nm[i];
        for (int i = 0; i < 16; ++i)  v[i] = nv[i];
    }
    for (int i = 0; i < 256; ++i) Mb[b * 256 + i] = M[i];
    for (int i = 0; i < 16; ++i)  vb[b * 16 + i]  = v[i];
}

// ---------------------------------------------------------------------------
// Kernel 4: per-pixel selected-leaf feature via (M_b, v_b), channel-normalize,
// 16->64 ReLU, 64->1 dot; partial sums into acc[b]. 4 blocks per batch.
// ---------------------------------------------------------------------------
__global__ __launch_bounds__(256)
void head_kernel(const float* __restrict__ f0, const float* __restrict__ Mb,
                 const float* __restrict__ vb, const float* __restrict__ m1w,
                 const float* __restrict__ m1b, const float* __restrict__ m2w,
                 float* __restrict__ acc)
{
    const int b = blockIdx.x >> 2;
    const int q = blockIdx.x & 3;

    __shared__ float sM[256], sv[16], s1w[1024], s1b[64], s2w[64];
    for (int i = threadIdx.x; i < 256; i += 256) sM[i] = Mb[b * 256 + i];
    if (threadIdx.x < 16) sv[threadIdx.x] = vb[b * 16 + threadIdx.x];
    for (int i = threadIdx.x; i < 1024; i += 256) s1w[i] = m1w[i];
    if (threadIdx.x < 64) { s1b[threadIdx.x] = m1b[threadIdx.x]; s2w[threadIdx.x] = m2w[threadIdx.x]; }
    __syncthreads();

    const float* fb = f0 + (size_t)b * VDIM * HW;
    float lsum = 0.f;
    for (int p = q * 1024 + threadIdx.x; p < (q + 1) * 1024; p += 256) {
        float x[16];
        #pragma unroll
        for (int c = 0; c < 16; ++c) x[c] = fb[c * HW + p];   // coalesced
        float y[16]; float n2 = 0.f;
        #pragma unroll
        for (int o = 0; o < 16; ++o) {
            float s = sv[o];
            for (int c = 0; c < 16; ++c) s += sM[o * 16 + c] * x[c];
            y[o] = s; n2 += s * s;
        }
        const float inv = 1.0f / (sqrtf(n2) + 1e-8f);
        #pragma unroll
        for (int o = 0; o < 16; ++o) y[o] *= inv;
        float ps = 0.f;
        for (int j = 0; j < 64; ++j) {
            float s = s1b[j];
            #pragma unroll
            for (int c = 0; c < 16; ++c) s += s1w[j * 16 + c] * y[c];
            ps += s2w[j] * fmaxf(s, 0.f);
        }
        lsum += ps;
    }

    __shared__ float red[256];
    red[threadIdx.x] = lsum;
    __syncthreads();
    for (int st = 128; st > 0; st >>= 1) {
        if (threadIdx.x < st) red[threadIdx.x] += red[threadIdx.x + st];
        __syncthreads();
    }
    if (threadIdx.x == 0) atomicAdd(&acc[b], red[0]);
}

// ---------------------------------------------------------------------------
// Kernel 5: mantissa = sigmoid(mean + m2_b) * 0.75 + 0.75
// ---------------------------------------------------------------------------
__global__ void final_kernel(const float* __restrict__ acc, const float* __restrict__ m2b,
                             float* __restrict__ dout)
{
    const int b = threadIdx.x;
    if (b < BATCH) {
        float raw = acc[b] * (1.0f / HW) + m2b[0];
        float sg = 1.0f / (1.0f + __expf(-raw));
        dout[160 + b] = sg * 0.75f + 0.75f;
    }
}

extern "C" void kernel_launch(void* const* d_in, const int* in_sizes, int n_in,
                              void* d_out, int out_size, void* d_ws, size_t ws_size,
                              hipStream_t stream)
{
    const float* features = (const float*)d_in[0];
    const float* root_w   = (const float*)d_in[1];
    const float* root_b   = (const float*)d_in[2];
    const float* level_w  = (const float*)d_in[3];
    const float* level_b  = (const float*)d_in[4];
    const float* m1_w     = (const float*)d_in[5];
    const float* m1_b     = (const float*)d_in[6];
    const float* m2_w     = (const float*)d_in[7];
    const float* m2_b     = (const float*)d_in[8];
    float* out = (float*)d_out;

    float* ws   = (float*)d_ws;
    float* f0   = ws;                                   // 32*16*4096 = 2,097,152 floats
    float* Mb   = f0 + (size_t)BATCH * VDIM * HW;       // 32*256
    float* vb   = Mb + BATCH * 256;                     // 32*16
    float* mean = vb + BATCH * 16;                      // 32*16
    float* acc  = mean + BATCH * 16;                    // 32

    hipMemsetAsync(acc, 0, BATCH * sizeof(float), stream);

    root_conv_wmma<<<BATCH * 8, 256, 0, stream>>>(features, root_w, root_b, f0);
    mean_kernel<<<BATCH * VDIM, 256, 0, stream>>>(f0, mean);
    tree_kernel<<<1, 32, 0, stream>>>(mean, level_w, level_b, Mb, vb, out);
    head_kernel<<<BATCH * 4, 256, 0, stream>>>(f0, Mb, vb, m1_w, m1_b, m2_w, acc);
    final_kernel<<<1, 32, 0, stream>>>(acc, m2_b, out);
}